// Index_29111288332314
// MI455X (gfx1250) — compile-verified
//
#include <hip/hip_runtime.h>
#include <hip/hip_bf16.h>

typedef __attribute__((ext_vector_type(2))) float v2f;
typedef __attribute__((ext_vector_type(8))) float v8f;

#define D_DIM 512   // feature dimension (fixed by the reference problem)
#define Q_MAX 2048  // queries (fixed by the reference problem)

// ---------------------------------------------------------------------------
// Kernel 1: WMMA f32 GEMM producing only the 32 surviving columns.
//   dcol[j][q] = sum_d index[(N-32)+j][d] * query[q][d]
// One wave computes one 16x16 (q x j) tile via V_WMMA_F32_16X16X4_F32,
// sweeping K = 512 in steps of 4.
//
// Fragment layouts (CDNA5 ISA 7.12.2, 32-bit, wave32):
//   A (16x4, MxK): lane L<16 -> row M=L, holds K=0 (v0), K=1 (v1);
//                  lane L>=16 -> row M=L-16, holds K=2 (v0), K=3 (v1).
//   B (4x16, KxN): mirrored: lane L<16 -> col N=L, rows K=0/1;
//                  lane L>=16 -> col N=L-16, rows K=2/3.
//   C/D (16x16 f32, 8 VGPRs): vgpr r, lane L -> M = r + 8*(L>=16), N = L%16.
// ---------------------------------------------------------------------------
__global__ void knn_gemm_wmma(const float* __restrict__ query,
                              const float* __restrict__ index,
                              float* __restrict__ dcol,
                              int Q, int Nrows, int Kcols) {
    const int gthread = blockIdx.x * blockDim.x + threadIdx.x;
    const int wave    = gthread >> 5;
    const int lane    = threadIdx.x & 31;

    const int jtiles = Kcols >> 4;          // 32/16 = 2 column-tiles
    const int qt = wave / jtiles;           // q-tile index
    const int jt = wave - qt * jtiles;      // j-tile index
    const int qbase   = qt << 4;
    const int rowbase = Nrows - Kcols + (jt << 4);  // first index-row of tile

    const int m  = lane & 15;               // row (A) / col (B) within tile
    const int hi = lane >> 4;               // lane half selects K sub-pair
    const int kb = hi << 1;                 // K offset {0,2} within each K=4 step

    const float* aptr = query + (size_t)(qbase + m)   * D_DIM + kb;
    const float* bptr = index + (size_t)(rowbase + m) * D_DIM + kb;

    v8f acc = {};
#pragma unroll 8
    for (int k0 = 0; k0 < D_DIM; k0 += 4) {
        v2f a = *(const v2f*)aptr;          // A[m][k0+kb], A[m][k0+kb+1]
        v2f b = *(const v2f*)bptr;          // B[k0+kb][n], B[k0+kb+1][n]
        aptr += 4;
        bptr += 4;
        // D = A(16x4) * B(4x16) + C, full f32 precision
        acc = __builtin_amdgcn_wmma_f32_16x16x4_f32(
            /*neg_a=*/false, a, /*neg_b=*/false, b,
            /*c_mod=*/(short)0, acc, /*reuse_a=*/false, /*reuse_b=*/false);
    }

    // Scatter the 16x16 f32 D tile into column-major workspace dcol[Kcols][Q]
    const int n = m;                        // D column (lane % 16)
    float* out = dcol + (size_t)((jt << 4) + n) * Q + qbase + (hi << 3);
#pragma unroll
    for (int r = 0; r < 8; ++r) out[r] = acc[r];
}

// ---------------------------------------------------------------------------
// Kernel 2: per-column bitonic key/index sort in LDS (ascending, stable via
// index tie-break), matching jnp.sort / jnp.argsort along axis 0.
// One 1024-thread workgroup sorts one column of 2048 entries.
// ---------------------------------------------------------------------------
__global__ void knn_sort_cols(const float* __restrict__ dcol,
                              float* __restrict__ out_vals,
                              int* __restrict__ out_idx,
                              int Q, int Kcols) {
    __shared__ float skey[Q_MAX];
    __shared__ int   sidx[Q_MAX];

    const int col = blockIdx.x;
    const float* cptr = dcol + (size_t)col * Q;

    for (int i = threadIdx.x; i < Q; i += blockDim.x) {
        skey[i] = cptr[i];
        sidx[i] = i;
    }
    __syncthreads();

    // Bitonic sort: direction per block of `size` from (lo & size); the final
    // merge (size == Q) is globally ascending.
    for (int size = 2; size <= Q; size <<= 1) {
        for (int stride = size >> 1; stride > 0; stride >>= 1) {
            for (int t = threadIdx.x; t < (Q >> 1); t += blockDim.x) {
                const int lo = 2 * t - (t & (stride - 1));
                const int hj = lo + stride;
                const bool asc = ((lo & size) == 0);
                const float ka = skey[lo], kc = skey[hj];
                const int   ia = sidx[lo], ic = sidx[hj];
                const bool gt = (ka > kc) || (ka == kc && ia > ic);
                if (gt == asc) {
                    skey[lo] = kc; skey[hj] = ka;
                    sidx[lo] = ic; sidx[hj] = ia;
                }
            }
            __syncthreads();
        }
    }

    // Output row-major [Q][Kcols]: values then int32 indices.
    for (int i = threadIdx.x; i < Q; i += blockDim.x) {
        out_vals[(size_t)i * Kcols + col] = skey[i];
        out_idx [(size_t)i * Kcols + col] = sidx[i];
    }
}

extern "C" void kernel_launch(void* const* d_in, const int* in_sizes, int n_in,
                              void* d_out, int out_size, void* d_ws, size_t ws_size,
                              hipStream_t stream) {
    const float* index = (const float*)d_in[0];   // [N, 512]
    const float* query = (const float*)d_in[1];   // [Q, 512]
    (void)n_in; (void)ws_size; (void)out_size;

    const int D     = D_DIM;
    const int Nrows = in_sizes[0] / D;            // 131072
    const int Q     = in_sizes[1] / D;            // 2048
    const int Kcols = 32;                         // k (reference scalar) = 32

    float* dcol = (float*)d_ws;                   // [Kcols][Q] f32 = 256 KB

    // GEMM: one wave per 16x16 tile -> (Q/16)*(Kcols/16) waves, 4 waves/block
    const int waves   = (Q >> 4) * (Kcols >> 4);
    const int threads = 128;
    const int blocks  = (waves * 32 + threads - 1) / threads;
    knn_gemm_wmma<<<dim3(blocks), dim3(threads), 0, stream>>>(
        query, index, dcol, Q, Nrows, Kcols);

    // Sort: one workgroup per column
    float* out_vals = (float*)d_out;
    int*   out_idx  = (int*)((float*)d_out + (size_t)Q * Kcols);
    knn_sort_cols<<<dim3(Kcols), dim3(1024), 0, stream>>>(
        dcol, out_vals, out_idx, Q, Kcols);
}